// ChamferLoss_28475633173109
// MI455X (gfx1250) — compile-verified
//
#include <hip/hip_runtime.h>
#include <math.h>

typedef float v2f __attribute__((ext_vector_type(2)));
typedef float v8f __attribute__((ext_vector_type(8)));

#define NPTS   4096
#define NBATCH 8
#define NPLANE 3
#define NPAIR  (NBATCH * NPLANE)        // 24
#define ROWS_PER_BLOCK 128
#define GRID_Y (NPTS / ROWS_PER_BLOCK)  // 32
#define NTILE  (NPTS / 16)              // 256
#define INF_BITS 0x7F800000u
#define REG_COEF 25.0f

// ---------------------------------------------------------------------------
// Workspace layout:
//   offset 0   : float  rowsum accumulator
//   offset 16  : uint   colmin[NPAIR * NPTS]  (float bit patterns, all >= 0)
// ---------------------------------------------------------------------------

__global__ void init_ws_kernel(float* accum, unsigned* colmin) {
    int t = blockIdx.x * blockDim.x + threadIdx.x;
    if (t == 0) *accum = 0.0f;
    if (t < NPAIR * NPTS) colmin[t] = INF_BITS;
}

__global__ __launch_bounds__(256) void chamfer_wmma_kernel(
        const float* __restrict__ pts,     // (8, 4096, 3)
        const float* __restrict__ ypred,   // (8, 3, 4)
        float* __restrict__ accum,
        unsigned* __restrict__ gcol) {
    // Reflected points padded to stride 4: {yx, yy, yz, ||y||^2}  (64 KB)
    __shared__ float    ly[NPTS * 4];
    __shared__ unsigned lcol[NPTS];        // per-column min bits (16 KB)

    const int p = blockIdx.x;              // pair index 0..23
    const int b = p / NPLANE;
    const int h = p % NPLANE;
    const float* x = pts + (size_t)b * NPTS * 3;
    const int tid = threadIdx.x;

    // ---- plane: normalize normal, raw offset ----
    float nx = ypred[(b * NPLANE + h) * 4 + 0];
    float ny = ypred[(b * NPLANE + h) * 4 + 1];
    float nz = ypred[(b * NPLANE + h) * 4 + 2];
    float off = ypred[(b * NPLANE + h) * 4 + 3];
    float inv = rsqrtf(nx * nx + ny * ny + nz * nz);
    nx *= inv; ny *= inv; nz *= inv;

    // ---- init block colmin + reflect all points into LDS (with ||y||^2) ----
    for (int t = tid; t < NPTS; t += 256) lcol[t] = INF_BITS;
    for (int j = tid; j < NPTS; j += 256) {
        float px = x[j * 3 + 0], py = x[j * 3 + 1], pz = x[j * 3 + 2];
        float dist = nx * px + ny * py + nz * pz + off;
        float ryx = px - 2.0f * dist * nx;
        float ryy = py - 2.0f * dist * ny;
        float ryz = pz - 2.0f * dist * nz;
        float4 v;
        v.x = ryx; v.y = ryy; v.z = ryz;
        v.w = ryx * ryx + ryy * ryy + ryz * ryz;
        *(float4*)&ly[j * 4] = v;
    }
    __syncthreads();

    const int wave = tid >> 5;
    const int lane = tid & 31;
    const int half = lane >> 4;     // wave32: lanes 0-15 vs 16-31
    const int lrow = lane & 15;
    const int ibase = blockIdx.y * ROWS_PER_BLOCK + wave * 16;

    // ---- A fragment: 16x4 f32, A[m][k] = -2 * x_m[k], K=3 padded with 0 ----
    // lanes 0-15: M=lane, v0=K0, v1=K1 ; lanes 16-31: M=lane-16, v0=K2, v1=K3
    const int ia = ibase + lrow;
    float ax = x[ia * 3 + 0], ay = x[ia * 3 + 1], az = x[ia * 3 + 2];
    v2f afrag;
    afrag.x = half ? (-2.0f * az) : (-2.0f * ax);
    afrag.y = half ? 0.0f : (-2.0f * ay);

    // ---- per-lane xsq vector matching the C layout (VGPR r -> M = r + 8*half)
    v8f xsq;
#pragma unroll
    for (int k = 0; k < 8; ++k) {
        int m = ibase + k + 8 * half;
        float xx = x[m * 3 + 0], xy = x[m * 3 + 1], xz = x[m * 3 + 2];
        xsq[k] = xx * xx + xy * xy + xz * xz;
    }

    v8f rowmin;
#pragma unroll
    for (int k = 0; k < 8; ++k) rowmin[k] = __builtin_inff();

    // ---- main loop: one WMMA per 16x16 tile, y-tile load software-pipelined
    const float4* __restrict__ ly4 = (const float4*)ly;
    float4 yv = ly4[lrow];                 // prefetch tile 0
    for (int jt = 0; jt < NTILE; ++jt) {
        const float4 cur = yv;
        yv = ly4[(((jt + 1) & (NTILE - 1)) << 4) + lrow];  // prefetch next

        // B fragment: 4x16 f32. lanes 0-15: N=lane, v0=K0, v1=K1 ;
        // lanes 16-31: N=lane-16, v0=K2, v1=K3 (K3 = 0 pad)
        v2f bfrag;
        bfrag.x = half ? cur.z : cur.x;
        bfrag.y = half ? 0.0f : cur.y;

        v8f c = {};
        c = __builtin_amdgcn_wmma_f32_16x16x4_f32(
                false, afrag, false, bfrag, (short)0, c, false, false);

        const float ysq = cur.w;
        float cp = __builtin_inff();
#pragma unroll
        for (int k = 0; k < 8; ++k) {
            float d = c[k] + xsq[k] + ysq;      // = ||x_m - y_j||^2
            rowmin[k] = fminf(rowmin[k], d);
            cp = fminf(cp, d);
        }
        // column min: lanes L and L+16 hit the same column address; the LDS
        // atomic resolves both halves and all waves (uint order == float
        // order for non-negative floats).
        atomicMin(&lcol[(jt << 4) + lrow], __float_as_uint(cp));
    }

    // ---- reduce row mins across the 16 lanes of each half, sum 16 rows ----
    float rsum = 0.0f;
#pragma unroll
    for (int k = 0; k < 8; ++k) {
        float r = rowmin[k];
        r = fminf(r, __shfl_xor(r, 1, 32));
        r = fminf(r, __shfl_xor(r, 2, 32));
        r = fminf(r, __shfl_xor(r, 4, 32));
        r = fminf(r, __shfl_xor(r, 8, 32));
        rsum += r;   // lane 0: rows M=0..7 ; lane 16: rows M=8..15
    }
    float osum = __shfl_xor(rsum, 16, 32);
    if (lane == 0) atomicAdd(accum, rsum + osum);

    // ---- flush block-local column mins to the global per-pair array ----
    __syncthreads();
    unsigned* gc = gcol + (size_t)p * NPTS;
    for (int t = tid; t < NPTS; t += 256) atomicMin(&gc[t], lcol[t]);
}

__global__ __launch_bounds__(256) void finalize_kernel(
        const float* __restrict__ ypred,
        const float* __restrict__ accum,
        const unsigned* __restrict__ gcol,
        float* __restrict__ out) {
    __shared__ float red[256];
    const int tid = threadIdx.x;
    float s = 0.0f;
    for (int t = tid; t < NPAIR * NPTS; t += 256) s += __uint_as_float(gcol[t]);
    red[tid] = s;
    __syncthreads();
    for (int stride = 128; stride > 0; stride >>= 1) {
        if (tid < stride) red[tid] += red[tid + stride];
        __syncthreads();
    }
    if (tid == 0) {
        float reg = 0.0f;
        for (int bb = 0; bb < NBATCH; ++bb) {
            float n[NPLANE][3];
            for (int hh = 0; hh < NPLANE; ++hh) {
                float vx = ypred[(bb * NPLANE + hh) * 4 + 0];
                float vy = ypred[(bb * NPLANE + hh) * 4 + 1];
                float vz = ypred[(bb * NPLANE + hh) * 4 + 2];
                float iv = rsqrtf(vx * vx + vy * vy + vz * vz);
                n[hh][0] = vx * iv; n[hh][1] = vy * iv; n[hh][2] = vz * iv;
            }
            float fro = 0.0f;
            for (int r = 0; r < NPLANE; ++r)
                for (int c = 0; c < NPLANE; ++c) {
                    float g = n[r][0] * n[c][0] + n[r][1] * n[c][1] +
                              n[r][2] * n[c][2] - (r == c ? 1.0f : 0.0f);
                    fro += g * g;
                }
            reg += sqrtf(fro);
        }
        out[0] = red[0] + *accum + REG_COEF * reg;
    }
}

extern "C" void kernel_launch(void* const* d_in, const int* in_sizes, int n_in,
                              void* d_out, int out_size, void* d_ws, size_t ws_size,
                              hipStream_t stream) {
    const float* pts   = (const float*)d_in[0];   // (8, 4096, 3) f32
    const float* ypred = (const float*)d_in[1];   // (8, 3, 4)  f32
    float*    accum = (float*)d_ws;
    unsigned* gcol  = (unsigned*)((char*)d_ws + 16);

    init_ws_kernel<<<(NPAIR * NPTS + 255) / 256, 256, 0, stream>>>(accum, gcol);
    dim3 grid(NPAIR, GRID_Y);
    chamfer_wmma_kernel<<<grid, 256, 0, stream>>>(pts, ypred, accum, gcol);
    finalize_kernel<<<1, 256, 0, stream>>>(ypred, accum, gcol, (float*)d_out);
}